// MiniRocketFeatures_48430051229775
// MI455X (gfx1250) — compile-verified
//
#include <hip/hip_runtime.h>
#include <hip/hip_bf16.h>
#include <stdint.h>

typedef float v2f __attribute__((ext_vector_type(2)));
typedef float v8f __attribute__((ext_vector_type(8)));

#define NKER 10000
#define B_   256
#define C_   23
#define L_   4096
#define NGRP 12

// ---------------------------------------------------------------------------
// K0: deterministic config (stand-in for np.random.default_rng(0); PCG64 +
// SeedSequence is not reproducible in-kernel). Same structure: k in {7,9},
// dilation = 2^[0..5], bias uniform [-1,1).
// ---------------------------------------------------------------------------
__device__ __forceinline__ uint64_t splitmix64(uint64_t z) {
    z += 0x9E3779B97F4A7C15ull;
    z = (z ^ (z >> 30)) * 0xBF58476D1CE4E5B9ull;
    z = (z ^ (z >> 27)) * 0x94D049BB133111EBull;
    return z ^ (z >> 31);
}

__global__ void cfg_init(int* __restrict__ klen, int* __restrict__ dil,
                         float* __restrict__ bias) {
    int i = blockIdx.x * blockDim.x + threadIdx.x;
    if (i >= NKER) return;
    uint64_t r = splitmix64(0xcbf29ce484222325ull + (uint64_t)i);
    klen[i] = (r & 1u) ? 9 : 7;
    dil[i]  = 1 << (int)((r >> 8) % 6u);
    uint32_t u = (uint32_t)(r >> 32);
    bias[i] = (float)u * (2.0f / 4294967296.0f) - 1.0f;
}

// ---------------------------------------------------------------------------
// K1: s[b,l] = sum_c x[b,c,l] via V_WMMA_F32_16X16X4_F32 (exact f32 math).
// A = ones(16x4). B tile = 4 channels x 16 positions:
//   B VGPR0 = x[c0+2*hi, col], VGPR1 = x[c0+2*hi+1, col]  (hi = lane>>4)
// Two independent accumulator chains (two 16-col tiles) per wave -> lanes
// 0..15 emit tile0 row M=0, lanes 16..31 emit tile1 row M=8; the wave stores
// one contiguous 128B chunk of s. Grid sized exactly: EXEC all-ones (WMMA req).
// This kernel is the HBM-bound pass: 96.5 MB read ~= 4.1 us at 23.3 TB/s.
// ---------------------------------------------------------------------------
__global__ __launch_bounds__(256) void chansum_wmma(const float* __restrict__ x,
                                                    float* __restrict__ s) {
    const int lane = threadIdx.x & 31;
    const int wave = (int)((blockIdx.x * blockDim.x + threadIdx.x) >> 5);
    const int b  = wave >> 7;           // 128 waves per batch row
    const int l0 = (wave & 127) << 5;   // 32 columns per wave
    const int hi = lane >> 4;
    const int li = lane & 15;

    v2f a; a.x = 1.0f; a.y = 1.0f;      // ones 16x4 A-matrix
    v8f acc0 = {};
    v8f acc1 = {};

    const float* xb = x + (size_t)b * C_ * L_;
    #pragma unroll
    for (int c0 = 0; c0 < 24; c0 += 4) {
        const int ca = c0 + 2 * hi;
        const int cb = ca + 1;
        // software-prefetch next channel group (strided walk defeats HW pf)
        if (c0 + 4 < 24) {
            const int cn = c0 + 4 + 2 * hi;
            if (cn < C_) __builtin_prefetch(&xb[(size_t)cn * L_ + l0 + li], 0, 1);
        }
        const float a0 = (ca < C_) ? xb[(size_t)ca * L_ + l0 + li]      : 0.0f;
        const float a1 = (cb < C_) ? xb[(size_t)cb * L_ + l0 + li]      : 0.0f;
        const float b0 = (ca < C_) ? xb[(size_t)ca * L_ + l0 + 16 + li] : 0.0f;
        const float b1 = (cb < C_) ? xb[(size_t)cb * L_ + l0 + 16 + li] : 0.0f;
        v2f B0; B0.x = a0; B0.y = a1;
        v2f B1; B1.x = b0; B1.y = b1;
        acc0 = __builtin_amdgcn_wmma_f32_16x16x4_f32(false, a, false, B0,
                                                     (short)0, acc0, false, false);
        acc1 = __builtin_amdgcn_wmma_f32_16x16x4_f32(false, a, false, B1,
                                                     (short)0, acc1, false, false);
    }
    const float v = hi ? acc1[0] : acc0[0];   // D VGPR0: M=0/N=li | M=8/N=li
    s[(size_t)b * L_ + l0 + lane] = v;
}

// ---------------------------------------------------------------------------
// K2: one block per (group g, batch b). Load s[b,:] into LDS once (padded by
// 128 zeros each side; max pad = 8*32/2 = 128), compute the dilated ones-conv
// (16 values/thread, kept in registers), block-max, and a 4-pass radix select
// on monotonic uint keys for sorted ranks 1351 and 2703.
//   q66 - q33 > 0  <=>  v_sorted[2703] > v_sorted[1351]   (exact, see analysis)
// ---------------------------------------------------------------------------
__global__ __launch_bounds__(256) void group_stats(const float* __restrict__ s,
                                                   float* __restrict__ m_tab,
                                                   float* __restrict__ f2_tab) {
    const int g   = blockIdx.x;   // 0..11
    const int b   = blockIdx.y;   // 0..255
    const int tid = threadIdx.x;
    const int klen = (g < 6) ? 7 : 9;
    const int d    = 1 << (g % 6);
    const int pad  = (klen - 1) * d / 2;

    __shared__ float    sp[L_ + 256];
    __shared__ uint32_t hist[256];
    __shared__ uint32_t cum[256];
    __shared__ float    redf[256];
    __shared__ uint32_t bc_prefix, bc_rank;

    if (tid < 128) { sp[tid] = 0.0f; sp[128 + L_ + tid] = 0.0f; }
    const float* sb = s + (size_t)b * L_;
    for (int l = tid; l < L_; l += 256) sp[128 + l] = sb[l];
    __syncthreads();

    uint32_t keys[16];
    float vmax = -3.402823466e38f;
    #pragma unroll
    for (int j = 0; j < 16; ++j) {
        const int p = tid + j * 256;
        const float* base = &sp[128 + p - pad];
        float acc = 0.0f;
        for (int k = 0; k < klen; ++k) acc += base[k * d];
        vmax = fmaxf(vmax, acc);
        const uint32_t u = __float_as_uint(acc);
        keys[j] = (u & 0x80000000u) ? ~u : (u | 0x80000000u);  // order-preserving
    }

    // block max
    redf[tid] = vmax; __syncthreads();
    for (int off = 128; off > 0; off >>= 1) {
        if (tid < off) redf[tid] = fmaxf(redf[tid], redf[tid + off]);
        __syncthreads();
    }
    const float m = redf[0];
    __syncthreads();

    // radix select (MSB->LSB, 8 bits/pass) for the two ranks
    uint32_t selkey[2];
    const int ranks[2] = {1351, 2703};
    for (int ri = 0; ri < 2; ++ri) {
        uint32_t prefix = 0;
        uint32_t rank   = (uint32_t)ranks[ri];
        for (int shift = 24; shift >= 0; shift -= 8) {
            hist[tid] = 0; __syncthreads();
            const uint32_t hm = (shift == 24) ? 0u : (0xFFFFFFFFu << (shift + 8));
            #pragma unroll
            for (int j = 0; j < 16; ++j) {
                if ((keys[j] & hm) == prefix)
                    atomicAdd(&hist[(keys[j] >> shift) & 255u], 1u);
            }
            __syncthreads();
            // Hillis-Steele inclusive scan over 256 bins
            const uint32_t cv = hist[tid];
            cum[tid] = cv; __syncthreads();
            for (int off = 1; off < 256; off <<= 1) {
                const uint32_t t = (tid >= off) ? cum[tid - off] : 0u;
                __syncthreads();
                cum[tid] += t;
                __syncthreads();
            }
            const uint32_t inc = cum[tid];
            const uint32_t exc = inc - cv;
            if (inc > rank && exc <= rank) {   // exactly one bin qualifies
                bc_prefix = prefix | ((uint32_t)tid << shift);
                bc_rank   = rank - exc;
            }
            __syncthreads();
            prefix = bc_prefix;
            rank   = bc_rank;
            __syncthreads();
        }
        selkey[ri] = prefix;
    }

    if (tid == 0) {
        m_tab[b * NGRP + g]  = m;
        f2_tab[b * NGRP + g] = (selkey[1] > selkey[0]) ? 1.0f : 0.0f;
    }
}

// ---------------------------------------------------------------------------
// K3: scatter per-kernel features. Each thread writes one interleaved
// (f1, f2) pair as a non-temporal native float2 vector (20 MB write-once).
// ---------------------------------------------------------------------------
__global__ __launch_bounds__(256) void scatter_out(const int* __restrict__ klen,
                                                   const int* __restrict__ dil,
                                                   const float* __restrict__ bias,
                                                   const float* __restrict__ m_tab,
                                                   const float* __restrict__ f2_tab,
                                                   v2f* __restrict__ out) {
    const int i = blockIdx.x * blockDim.x + threadIdx.x;
    const int b = blockIdx.y;
    if (i >= NKER) return;
    const int e = 31 - __clz(dil[i]);                 // log2(dilation)
    const int g = ((klen[i] == 9) ? 6 : 0) + e;
    const float m  = m_tab[b * NGRP + g];
    v2f o;
    o.x = (m > bias[i]) ? 1.0f : 0.0f;
    o.y = f2_tab[b * NGRP + g];
    __builtin_nontemporal_store(o, &out[(size_t)b * NKER + i]);
}

// ---------------------------------------------------------------------------
// Host launcher. ws layout: s (4 MB) | m_tab (12 KB) | f2_tab (12 KB) |
// klen (40 KB) | dil (40 KB) | bias (40 KB)  ~= 4.35 MB total.
// ---------------------------------------------------------------------------
extern "C" void kernel_launch(void* const* d_in, const int* in_sizes, int n_in,
                              void* d_out, int out_size, void* d_ws, size_t ws_size,
                              hipStream_t stream) {
    const float* x = (const float*)d_in[0];
    char* ws = (char*)d_ws;
    const size_t S_BYTES = (size_t)B_ * L_ * sizeof(float);   // 4 MB
    float* s      = (float*)(ws);
    float* m_tab  = (float*)(ws + S_BYTES);
    float* f2_tab = (float*)(ws + S_BYTES + 16384);
    int*   klen   = (int*)  (ws + S_BYTES + 32768);
    int*   dil    = (int*)  (ws + S_BYTES + 32768 + 40000);
    float* bias   = (float*)(ws + S_BYTES + 32768 + 80000);

    cfg_init<<<(NKER + 255) / 256, 256, 0, stream>>>(klen, dil, bias);

    // 256 batches * 128 waves = 32768 waves, 8 waves/block -> 4096 blocks
    chansum_wmma<<<4096, 256, 0, stream>>>(x, s);

    dim3 g2(NGRP, B_);
    group_stats<<<g2, 256, 0, stream>>>(s, m_tab, f2_tab);

    dim3 g3((NKER + 255) / 256, B_);
    scatter_out<<<g3, 256, 0, stream>>>(klen, dil, bias, m_tab, f2_tab,
                                        (v2f*)d_out);
}